// EmbeddingLoss_59382217834933
// MI455X (gfx1250) — compile-verified
//
#include <hip/hip_runtime.h>
#include <math.h>

typedef __attribute__((ext_vector_type(2))) float v2f;
typedef __attribute__((ext_vector_type(4))) float v4f;
typedef __attribute__((ext_vector_type(8))) float v8f;

#define NB   8        // batch
#define NF   64       // features
#define NC   16       // codes
#define NL   65536    // h*w = 256*256
#define EPSV 1e-6f
#define MARG 1.0f

// Workspace layout (floats):
//   [0      .. 255]  crossAcc[call][b][c]   (2*8*16)
//   [256    .. 511]  rsumAcc [call][b][c]   (2*8*16)

// Fused streaming kernel: per 16-pixel tile, compute
//   dot[m][n] = sum_f e[f][m] * cb[f][n]   via 16x V_WMMA_F32_16X16X4_F32
//   ee[m]     = sum_f e[f][m]^2            (byproduct of A-fragment loads)
//   acc_sq[n] += (ee[m] - 2*dot[m][n] + cbsq[n]) * r[n][m],  acc_r[n] += r[n][m]
// Streaming data (embed, r) is read exactly once and exceeds the 192MB L2,
// so those loads are non-temporal; the codebook (reused by all waves) stays RT.
__global__ __launch_bounds__(256) void cross_partial_kernel(
    const float* __restrict__ e1, const float* __restrict__ r2,
    const float* __restrict__ e2, const float* __restrict__ r1,
    const float* __restrict__ cb, float* __restrict__ acc)
{
    const int call = blockIdx.z;            // 0: (e1, r2)   1: (e2, r1)
    const int b    = blockIdx.y;
    const int bx   = blockIdx.x;            // 0..127
    const int tid  = threadIdx.x;
    const int wave = tid >> 5;              // 0..7
    const int lane = tid & 31;
    const int half = lane >> 4;             // lane-half (K group select)
    const int lp   = lane & 15;             // pixel (A/ee) or code (B/acc) index

    const float* e = (call == 0) ? e1 : e2;
    const float* r = (call == 0) ? r2 : r1;
    const float* eb = e + (size_t)b * NF * NL;
    const float* rb = r + (size_t)b * NC * NL;

    // ---- codebook B fragments, resident for the whole wave ----
    // B (4xK rows x 16 cols) per step kk: lane-half h holds K = 2h+{0,1} in {v0,v1},
    // col n = lane&15 (mirrors the fp32 A/C layouts of ISA 7.12.2).
    v2f bmat[16];
    float cbsq_part = 0.f;
#pragma unroll
    for (int kk = 0; kk < 16; ++kk) {
        float b0 = cb[(4 * kk + 2 * half + 0) * NC + lp];
        float b1 = cb[(4 * kk + 2 * half + 1) * NC + lp];
        bmat[kk].x = b0;
        bmat[kk].y = b1;
        cbsq_part += b0 * b0 + b1 * b1;
    }
    // combine the two K-halves -> full sum_f cb[f][n]^2 on every lane
    float cbsq = cbsq_part + __shfl_xor(cbsq_part, 16, 32);

    float acc_sq = 0.f, acc_r = 0.f;

#pragma unroll 1
    for (int t = 0; t < 4; ++t) {
        const int base = ((bx * 8 + wave) * 4 + t) << 4;   // first of 16 pixels

        v8f  cfrag = {};          // dot accumulator (16x16 fp32)
        float ee_part = 0.f;
#pragma unroll
        for (int kk = 0; kk < 16; ++kk) {
            // A fragment: pixel m = lane&15, features K = 4kk + 2h + {0,1}
            float a0 = __builtin_nontemporal_load(
                eb + (size_t)(4 * kk + 2 * half + 0) * NL + base + lp);
            float a1 = __builtin_nontemporal_load(
                eb + (size_t)(4 * kk + 2 * half + 1) * NL + base + lp);
            v2f a; a.x = a0; a.y = a1;
            ee_part += a0 * a0 + a1 * a1;
            cfrag = __builtin_amdgcn_wmma_f32_16x16x4_f32(
                false, a, false, bmat[kk], (short)0, cfrag, false, false);
        }
        // full sum_f e^2 for pixel (lane&15), identical on both lane-halves
        float ee = ee_part + __shfl_xor(ee_part, 16, 32);

        // r[n][pixel m] for this lane's 8 rows (m = 8*half .. 8*half+7):
        // 8 consecutive floats, 32B-aligned -> two 16B non-temporal loads.
        const v4f* rp = (const v4f*)(rb + (size_t)lp * NL + base + 8 * half);
        v4f rv0 = __builtin_nontemporal_load(rp + 0);
        v4f rv1 = __builtin_nontemporal_load(rp + 1);

        // C layout: cfrag[j] = dot[m = j + 8*half][n = lane&15]
#pragma unroll
        for (int j = 0; j < 8; ++j) {
            int   m   = j + 8 * half;
            float eem = __shfl(ee, m, 32);                  // ee[m]
            float rv  = (j < 4) ? rv0[j & 3] : rv1[j & 3];
            float sq  = eem - 2.f * cfrag[j] + cbsq;
            acc_sq += sq * rv;
            acc_r  += rv;
        }
    }

    // fold the two lane-halves (pixels 0-7 vs 8-15), then one atomic per code
    acc_sq += __shfl_xor(acc_sq, 16, 32);
    acc_r  += __shfl_xor(acc_r, 16, 32);
    if (lane < 16) {
        float* crossAcc = acc + ((size_t)call * NB + b) * NC;
        float* rsumAcc  = acc + 2 * NB * NC + ((size_t)call * NB + b) * NC;
        atomicAdd(&crossAcc[lp], acc_sq);
        atomicAdd(&rsumAcc[lp],  acc_r);
    }
}

// Single-block epilogue: l_cross from accumulators + codebook-only losses.
__global__ __launch_bounds__(256) void finalize_kernel(
    const float* __restrict__ cb, const float* __restrict__ acc,
    float* __restrict__ out)
{
    __shared__ float s_sum[2], s_cnt[2], s_dist, s_reg;
    const int tid = threadIdx.x;
    if (tid < 2) { s_sum[tid] = 0.f; s_cnt[tid] = 0.f; }
    if (tid == 0) { s_dist = 0.f; s_reg = 0.f; }
    __syncthreads();

    // l_cross: 256 (call,b,c) entries, one per thread
    {
        int   call  = tid >> 7;
        int   idx   = tid & 127;
        float a     = acc[call * 128 + idx];
        float rs    = acc[256 + call * 128 + idx];
        float valid = (rs != 0.f) ? 1.f : 0.f;
        float val   = (rs != 0.f) ? a / (rs + EPSV) : 0.f;
        atomicAdd(&s_sum[call], val);
        atomicAdd(&s_cnt[call], valid);
    }
    // l_dist: one (i,j) pair per thread (diagonal included, as in reference)
    {
        int i = tid >> 4, j = tid & 15;
        float sq = 0.f;
        for (int f = 0; f < NF; ++f) {
            float d = cb[f * NC + i] - cb[f * NC + j];
            sq += d * d;
        }
        float dist  = (sq > 0.f) ? sqrtf(sq) : 0.f;
        float hinge = fmaxf(2.f * MARG - dist, 0.f);
        atomicAdd(&s_dist, hinge * hinge);
    }
    // l_reg: column norms
    if (tid < NC) {
        float sq = 0.f;
        for (int f = 0; f < NF; ++f) { float v = cb[f * NC + tid]; sq += v * v; }
        atomicAdd(&s_reg, sqrtf(sq));
    }
    __syncthreads();
    if (tid == 0) {
        out[0] = s_sum[0] / s_cnt[0] + s_sum[1] / s_cnt[1];
        out[1] = s_dist / (2.f * NC * (NC - 1));
        out[2] = s_reg / (float)NC;
    }
}

extern "C" void kernel_launch(void* const* d_in, const int* in_sizes, int n_in,
                              void* d_out, int out_size, void* d_ws, size_t ws_size,
                              hipStream_t stream) {
    const float* e1 = (const float*)d_in[0];
    const float* r1 = (const float*)d_in[1];
    const float* e2 = (const float*)d_in[2];
    const float* r2 = (const float*)d_in[3];
    const float* cb = (const float*)d_in[4];
    float* acc = (float*)d_ws;

    hipMemsetAsync(d_ws, 0, 512 * sizeof(float), stream);   // capture-safe

    dim3 grid(128, NB, 2), block(256);
    cross_partial_kernel<<<grid, block, 0, stream>>>(e1, r2, e2, r1, cb, acc);
    finalize_kernel<<<1, 256, 0, stream>>>(cb, acc, (float*)d_out);
}